// PeronaMalik_20461224198746
// MI455X (gfx1250) — compile-verified
//
#include <hip/hip_runtime.h>
#include <cstdint>

#ifndef __has_builtin
#define __has_builtin(x) 0
#endif

#if __has_builtin(__builtin_amdgcn_global_load_async_to_lds_b128)
#define PM_ASYNC 1
#else
#define PM_ASYNC 0
#endif

#define PM_TILE_H 8
#define PM_W 1024
#define PM_NT 256

typedef int   v4i __attribute__((ext_vector_type(4)));
typedef float v4f __attribute__((ext_vector_type(4)));

// ---- async global -> LDS staging (CDNA5 ASYNCcnt path) ----------------------
__device__ __forceinline__ void pm_g2lds_b128(const float* g, float* l) {
#if PM_ASYNC
  // signature (from clang diagnostic): (v4i AS1*, v4i AS3*, imm offset, imm cpol)
  __builtin_amdgcn_global_load_async_to_lds_b128(
      (__attribute__((address_space(1))) v4i*)(uintptr_t)g,
      (__attribute__((address_space(3))) v4i*)(uint32_t)(uintptr_t)l,
      0, 0);
#else
  *(v4f*)l = *(const v4f*)g;
#endif
}

__device__ __forceinline__ void pm_wait_async() {
#if PM_ASYNC
#if __has_builtin(__builtin_amdgcn_s_wait_asynccnt)
  __builtin_amdgcn_s_wait_asynccnt(0);
#else
  asm volatile("s_wait_asynccnt 0" ::: "memory");
#endif
#endif
}

// ---- Perona-Malik diffusion step -------------------------------------------
// grid: (H/PM_TILE_H, N), block: 256 threads (8 wave32)
__global__ __launch_bounds__(PM_NT) void
PeronaMalik_20461224198746_kernel(const float* __restrict__ img,
                                  float* __restrict__ out, int H) {
  __shared__ __align__(16) float tile[(PM_TILE_H + 2) * PM_W];  // 40 KB

  const int tid = threadIdx.x;
  const int n   = blockIdx.y;
  const int y0  = blockIdx.x * PM_TILE_H;
  const size_t plane = (size_t)H * PM_W;
  const float* base  = img + (size_t)n * plane;
  const int c0 = tid * 4;  // 4 floats (16 B) per thread per row

  // Stage TILE_H + 2 halo rows into LDS with row reflection at image edges.
#pragma unroll
  for (int r = 0; r < PM_TILE_H + 2; ++r) {
    int gy = y0 - 1 + r;
    gy = (gy < 0) ? 1 : gy;            // ReflectionPad2d: row -1 -> row 1
    gy = (gy >= H) ? (H - 2) : gy;     //                 row H  -> row H-2
    pm_g2lds_b128(base + (size_t)gy * PM_W + c0, &tile[r * PM_W + c0]);
  }
  pm_wait_async();
  __syncthreads();

  const float nk = -100.0f;  // -1 / DIFFUSION_RATE^2
  const float dt = 0.15f;    // DELTA_T
  float* outp = out + (size_t)n * plane + (size_t)y0 * PM_W + c0;

#pragma unroll
  for (int r = 0; r < PM_TILE_H; ++r) {
    const float* up = &tile[(r + 0) * PM_W];
    const float* cn = &tile[(r + 1) * PM_W];
    const float* dn = &tile[(r + 2) * PM_W];

    v4f u = *(const v4f*)(up + c0);
    v4f c = *(const v4f*)(cn + c0);
    v4f d = *(const v4f*)(dn + c0);
    // Column reflection: left of x=0 is x=1; right of x=1023 is x=1022.
    float lf = (c0 == 0)        ? cn[1]        : cn[c0 - 1];
    float rt = (c0 + 4 >= PM_W) ? cn[PM_W - 2] : cn[c0 + 4];

    float gx = u.x + d.x + lf  + c.y - 4.0f * c.x;
    float gy = u.y + d.y + c.x + c.z - 4.0f * c.y;
    float gz = u.z + d.z + c.y + c.w - 4.0f * c.z;
    float gw = u.w + d.w + c.z + rt  - 4.0f * c.w;

    v4f o;
    o.x = c.x + gx * __expf(nk * gx * gx) * dt;
    o.y = c.y + gy * __expf(nk * gy * gy) * dt;
    o.z = c.z + gz * __expf(nk * gz * gz) * dt;
    o.w = c.w + gw * __expf(nk * gw * gw) * dt;
    o.x = fminf(fmaxf(o.x, 0.0f), 1.0f);
    o.y = fminf(fmaxf(o.y, 0.0f), 1.0f);
    o.z = fminf(fmaxf(o.z, 0.0f), 1.0f);
    o.w = fminf(fmaxf(o.w, 0.0f), 1.0f);

    // Output is never re-read: bypass cache retention (TH=NT store).
    __builtin_nontemporal_store(o, (v4f*)(outp + (size_t)r * PM_W));
  }
}

extern "C" void kernel_launch(void* const* d_in, const int* in_sizes, int n_in,
                              void* d_out, int out_size, void* d_ws, size_t ws_size,
                              hipStream_t stream) {
  const float* img = (const float*)d_in[0];
  // d_in[1] is the 3x3 Laplace kernel; it is the fixed 5-point stencil, which
  // this kernel hardcodes (u + d + l + r - 4c), matching the reference.
  float* out = (float*)d_out;

  const int H = 1024;
  const int W = PM_W;
  const int N = in_sizes[0] / (H * W);  // 64 images

  dim3 grid(H / PM_TILE_H, N);          // (128, 64)
  PeronaMalik_20461224198746_kernel<<<grid, PM_NT, 0, stream>>>(img, out, H);
}